// GraphSAGEEncoder_46420006535375
// MI455X (gfx1250) — compile-verified
//
#include <hip/hip_runtime.h>

// GraphSAGE 2-layer encoder for MI455X (gfx1250, wave32).
//   layer: m = segment_mean(h[src] -> dst);  out = relu([h|m] @ W + b)
// Aggregation: L2-resident atomic scatter-add (25.6MB tables << 192MB L2).
// GEMM: V_WMMA_F32_16X16X4_F32, f32 in / f32 accum to match reference dtype.

#define NNODES 50000
#define NEDGES 800000
#define DIM    128

typedef __attribute__((ext_vector_type(2))) float v2f;
typedef __attribute__((ext_vector_type(8))) float v8f;

// ---------------------------------------------------------------- utilities
__global__ void zero_f32(float* __restrict__ p, int n) {
  int i = blockIdx.x * blockDim.x + threadIdx.x;
  int stride = gridDim.x * blockDim.x;
  for (; i < n; i += stride) p[i] = 0.0f;
}

__global__ void degree_kernel(const int* __restrict__ dst, float* __restrict__ deg) {
  int e = blockIdx.x * blockDim.x + threadIdx.x;
  if (e < NEDGES) unsafeAtomicAdd(&deg[dst[e]], 1.0f);
}

// deg -> 1/max(deg,1) in place (reused by both layers)
__global__ void rdeg_kernel(float* __restrict__ deg) {
  int i = blockIdx.x * blockDim.x + threadIdx.x;
  if (i < NNODES) deg[i] = 1.0f / fmaxf(deg[i], 1.0f);
}

// One wave (32 lanes) per edge; each lane moves 4 consecutive floats.
// 8 edges per 256-thread block.
__global__ __launch_bounds__(256) void scatter_kernel(
    const float* __restrict__ h, const int* __restrict__ src,
    const int* __restrict__ dst, float* __restrict__ msg) {
  int e = blockIdx.x * 8 + (threadIdx.x >> 5);
  int c = (threadIdx.x & 31) * 4;
  if (e >= NEDGES) return;
  int s = src[e];
  int d = dst[e];
  const float4 v = *(const float4*)(h + (size_t)s * DIM + c);
  float* o = msg + (size_t)d * DIM + c;
  unsafeAtomicAdd(o + 0, v.x);
  unsafeAtomicAdd(o + 1, v.y);
  unsafeAtomicAdd(o + 2, v.z);
  unsafeAtomicAdd(o + 3, v.w);
}

// ---------------------------------------------------------------- SAGE GEMM
// out[M,128] = relu([h | msg*rdeg] @ W + bias),  W is [256,128] row-major.
// Block = 8 waves; block owns a 16-row stripe, wave w owns columns [16w,16w+16).
// K = 256 in steps of 4 via V_WMMA_F32_16X16X4_F32.
//
// A (16x4 f32, 2 VGPRs): lanes 0-15 -> K = {k, k+1}, lanes 16-31 -> K = {k+2, k+3},
//                        row M = lane & 15 (constant per lane across the K loop).
// B (4x16 f32, 2 VGPRs): VGPR0 = rows K {k / k+2}, VGPR1 = rows K {k+1 / k+3},
//                        column N = lane & 15.
// C/D (16x16, 8 VGPRs):  VGPR i -> row i (lanes 0-15) / row i+8 (lanes 16-31).
__global__ __launch_bounds__(256) void sage_gemm(
    const float* __restrict__ h,     // [M,128] self features
    const float* __restrict__ msg,   // [M,128] neighbor sums
    const float* __restrict__ rdeg,  // [M] 1/max(deg,1)
    const float* __restrict__ W,     // [256,128]
    const float* __restrict__ bias,  // [128]
    float* __restrict__ out)         // [M,128]
{
  const int m0   = blockIdx.x * 16;
  const int wave = threadIdx.x >> 5;   // 0..7 -> N tile
  const int lane = threadIdx.x & 31;
  const int l15  = lane & 15;
  const int hi   = lane >> 4;          // 0: lanes 0-15, 1: lanes 16-31
  const int koff = hi * 2;
  const int row  = m0 + l15;
  const int nn   = wave * 16 + l15;

  const float r = rdeg[row];
  const float* __restrict__ ah = h   + (size_t)row * DIM;
  const float* __restrict__ am = msg + (size_t)row * DIM;

  v8f c = {};

  // K = 0..127 : self-feature half of the concat
  #pragma unroll 4
  for (int k0 = 0; k0 < 128; k0 += 4) {
    v2f a, b;
    a.x = ah[k0 + koff];
    a.y = ah[k0 + koff + 1];
    b.x = W[(size_t)(k0 + koff)     * DIM + nn];
    b.y = W[(size_t)(k0 + koff + 1) * DIM + nn];
    c = __builtin_amdgcn_wmma_f32_16x16x4_f32(false, a, false, b,
                                              (short)0, c, false, false);
  }
  // K = 128..255 : neighbor-mean half (mean folded in as per-row scale)
  #pragma unroll 4
  for (int k0 = 0; k0 < 128; k0 += 4) {
    v2f a, b;
    a.x = am[k0 + koff] * r;
    a.y = am[k0 + koff + 1] * r;
    b.x = W[(size_t)(128 + k0 + koff)     * DIM + nn];
    b.y = W[(size_t)(128 + k0 + koff + 1) * DIM + nn];
    c = __builtin_amdgcn_wmma_f32_16x16x4_f32(false, a, false, b,
                                              (short)0, c, false, false);
  }

  const float bn = bias[nn];
  #pragma unroll
  for (int i = 0; i < 8; ++i) {
    int ro = m0 + i + hi * 8;
    out[(size_t)ro * DIM + nn] = fmaxf(c[i] + bn, 0.0f);
  }
}

// ---------------------------------------------------------------- launcher
extern "C" void kernel_launch(void* const* d_in, const int* in_sizes, int n_in,
                              void* d_out, int out_size, void* d_ws, size_t ws_size,
                              hipStream_t stream) {
  const float* h  = (const float*)d_in[0];
  const int* esrc = (const int*)d_in[1];
  const int* edst = (const int*)d_in[2];
  const float* W1 = (const float*)d_in[3];
  const float* b1 = (const float*)d_in[4];
  const float* W2 = (const float*)d_in[5];
  const float* b2 = (const float*)d_in[6];
  float* out = (float*)d_out;

  // workspace layout (floats): rdeg[50176 aligned] | msg[50000*128] | h1[50000*128]
  float* ws   = (float*)d_ws;
  float* rdeg = ws;
  float* msg  = ws + 50176;
  float* h1   = msg + (size_t)NNODES * DIM;

  const int nMsg = NNODES * DIM;

  // degree (shared across layers) + zero msg accumulator
  zero_f32<<<2048, 256, 0, stream>>>(rdeg, 50176 + nMsg);
  degree_kernel<<<(NEDGES + 255) / 256, 256, 0, stream>>>(edst, rdeg);
  rdeg_kernel<<<(NNODES + 255) / 256, 256, 0, stream>>>(rdeg);

  // layer 1
  scatter_kernel<<<NEDGES / 8, 256, 0, stream>>>(h, esrc, edst, msg);
  sage_gemm<<<NNODES / 16, 256, 0, stream>>>(h, msg, rdeg, W1, b1, h1);

  // layer 2
  zero_f32<<<2048, 256, 0, stream>>>(msg, nMsg);
  scatter_kernel<<<NEDGES / 8, 256, 0, stream>>>(h1, esrc, edst, msg);
  sage_gemm<<<NNODES / 16, 256, 0, stream>>>(h1, msg, rdeg, W2, b2, out);
}